// Quanv_70892730188028
// MI455X (gfx1250) — compile-verified
//
#include <hip/hip_runtime.h>
#include <hip/hip_bf16.h>

#define KK   3
#define FF   4
#define NQ   9
#define HSZ  96
#define HOUT 94
#define CIN  3

#define ROW_F  (HSZ)          // 96 floats per staged row
#define CH_F   (KK * ROW_F)   // 288 floats per channel (3 contiguous rows)
#define TOT_F  (CIN * CH_F)   // 864 floats staged per block
#define TOT_V4 (TOT_F / 4)    // 216 float4 chunks
#define CH_V4  (CH_F / 4)     // 72 chunks per channel

#define PI_F 3.14159265358979323846f

__global__ __launch_bounds__(256)
void quanv_closed_form_kernel(const float* __restrict__ x,
                              const float* __restrict__ w,
                              float* __restrict__ out) {
    __shared__ float sbuf[TOT_F];   // staged rows -> in-place cos(pi*x)
    __shared__ float sA[NQ];        // cos(w_j)

    const int tid  = threadIdx.x;
    const int row0 = blockIdx.x;    // output row; needs input rows row0..row0+2

    // ---- Stage 3 channels x (3 contiguous rows x 96 cols) into LDS ----
    // gfx1250 async VMEM->LDS DMA (ASYNCcnt), 16B per lane, single predicated issue.
    if (tid < TOT_V4) {
        const int c   = tid / CH_V4;          // 0..2  (single divide, computed once)
        const int rem = tid - c * CH_V4;      // 0..71 (chunk within contiguous 3-row span)
        const unsigned goff = (unsigned)(c * (HSZ * HSZ * 4)   // channel stride (bytes)
                                         + row0 * (HSZ * 4)    // first staged row (bytes)
                                         + rem * 16);          // chunk offset (bytes)
        const unsigned lds_addr = (unsigned)(size_t)(&sbuf[tid * 4]);
        asm volatile("global_load_async_to_lds_b128 %0, %1, %2"
                     :: "v"(lds_addr), "v"(goff), "s"(x)
                     : "memory");
    }

    // cos(w_j) once per block (independent of the staged data)
    if (tid < NQ) sA[tid] = cosf(w[tid]);

#if __has_builtin(__builtin_amdgcn_s_wait_asynccnt)
    __builtin_amdgcn_s_wait_asynccnt(0);
#else
    asm volatile("s_wait_asynccnt 0" ::: "memory");
#endif
    __syncthreads();

    // ---- In-place transform: v -> cos(pi * v). Each element owned by one thread. ----
#pragma unroll
    for (int l = tid; l < TOT_F; l += 256) {
        sbuf[l] = cosf(PI_F * sbuf[l]);
    }
    __syncthreads();

    // ---- Closed-form expectation values per output pixel ----
    // z_q = cos(w_q) * cos(pi * x_q), q = ki*3 + kj over the 3x3 patch.
    // ev0 = z1..z8,  ev1 = z0 z1,  ev2 = z0 z1 z2,  ev3 = z0 z1 z2 z3.
    if (tid < HOUT) {
        const int j = tid;
        float acc0 = 0.f, acc1 = 0.f, acc2 = 0.f, acc3 = 0.f;
        const float a0 = sA[0], a1 = sA[1], a2 = sA[2];
        const float a3 = sA[3], a4 = sA[4], a5 = sA[5];
        const float a6 = sA[6], a7 = sA[7], a8 = sA[8];
#pragma unroll
        for (int c = 0; c < CIN; ++c) {
            const float* b = &sbuf[c * CH_F];
            const float z0 = a0 * b[0 * ROW_F + j + 0];
            const float z1 = a1 * b[0 * ROW_F + j + 1];
            const float z2 = a2 * b[0 * ROW_F + j + 2];
            const float z3 = a3 * b[1 * ROW_F + j + 0];
            const float z4 = a4 * b[1 * ROW_F + j + 1];
            const float z5 = a5 * b[1 * ROW_F + j + 2];
            const float z6 = a6 * b[2 * ROW_F + j + 0];
            const float z7 = a7 * b[2 * ROW_F + j + 1];
            const float z8 = a8 * b[2 * ROW_F + j + 2];
            const float p01 = z0 * z1;          // ev1
            const float e2  = p01 * z2;         // ev2
            const float e3  = e2 * z3;          // ev3
            float r = z1 * z2;                  // ev0 = z1..z8
            r *= z3; r *= z4; r *= z5; r *= z6; r *= z7; r *= z8;
            acc0 += r; acc1 += p01; acc2 += e2; acc3 += e3;
        }
        float4 o; o.x = acc0; o.y = acc1; o.z = acc2; o.w = acc3;
        // flat-reinterpret layout: index = (i*94 + j)*4 + f  (16B aligned)
        *(float4*)(out + ((size_t)row0 * HOUT + j) * FF) = o;
    }
}

extern "C" void kernel_launch(void* const* d_in, const int* in_sizes, int n_in,
                              void* d_out, int out_size, void* d_ws, size_t ws_size,
                              hipStream_t stream) {
    (void)in_sizes; (void)n_in; (void)out_size; (void)d_ws; (void)ws_size;
    const float* x = (const float*)d_in[0];   // [1,3,96,96]
    const float* w = (const float*)d_in[1];   // [1,9]
    float* out = (float*)d_out;               // [1,4,94,94] (flat reinterpret of [94,94,4])
    quanv_closed_form_kernel<<<dim3(HOUT), dim3(256), 0, stream>>>(x, w, out);
}